// RnnGruModel_v2_27573690040798
// MI455X (gfx1250) — compile-verified
//
#include <hip/hip_runtime.h>

// tanh-RNN: SEQ=512, BATCH=4096, IN_DIM=1, HID=30 (pad->32), out dim 1.
// One wave32 per 16-batch tile; state transposed through per-wave LDS;
// 16x v_wmma_f32_16x16x4_f32 per timestep; y-reduction via ds_swizzle SWAPX16.

#define SEQ   512
#define BATCH 4096
#define HID   30
#define LROW  34              // padded LDS row stride (floats) for bank spread
#define WPB   4               // waves per block

typedef float v2f __attribute__((ext_vector_type(2)));
typedef float v4f __attribute__((ext_vector_type(4)));
typedef float v8f __attribute__((ext_vector_type(8)));

__device__ __forceinline__ float fast_tanh(float x) {
#if __has_builtin(__builtin_amdgcn_tanhf)
  return __builtin_amdgcn_tanhf(x);
#elif __has_builtin(__builtin_amdgcn_tanh_f32)
  return __builtin_amdgcn_tanh_f32(x);
#else
  float ax = __builtin_fabsf(x);
  float t  = __builtin_amdgcn_exp2f(ax * -2.885390081777927f); // 2^{-2|x|*log2 e}
  float r  = (1.0f - t) * __builtin_amdgcn_rcpf(1.0f + t);
  return __builtin_copysignf(r, x);
#endif
}

// Exchange values between lane i and lane i^16 (SWAPX16 group-of-32 swizzle).
__device__ __forceinline__ float swap_half(float v) {
  int r = __builtin_amdgcn_ds_swizzle(__builtin_bit_cast(int, v), 0x401F);
  return __builtin_bit_cast(float, r);
}

__global__ __launch_bounds__(32 * WPB)
void rnn_tanh_wmma_kernel(const float* __restrict__ X,
                          const float* __restrict__ W_ih,
                          const float* __restrict__ W_hh,
                          const float* __restrict__ b_ih,
                          const float* __restrict__ b_hh,
                          const float* __restrict__ W_out,
                          const float* __restrict__ b_out,
                          float* __restrict__ Y) {
  __shared__ float lds[WPB * 16 * LROW];

  const int lane = threadIdx.x & 31;
  const int wav  = threadIdx.x >> 5;
  const int lc   = lane & 15;     // column-in-tile / row index
  const int half = lane >> 4;     // 0 or 1 (selects K pair / M-high)
  const int b0   = (blockIdx.x * WPB + wav) * 16;   // batch tile base

  float* hbuf = lds + wav * 16 * LROW;

  // h0 = 0
  for (int i = lane; i < 16 * LROW; i += 32) hbuf[i] = 0.0f;

  // ---- Loop-invariant per-lane constants -------------------------------
  // B fragments: B(k,n) = W_hh[n][k]  (so A x B = h @ W_hh^T)
  // f32 4x16 B layout: lane = n%16 + 16*(k/2), VGPR = k%2
  v2f Bf[2][8];
  for (int t = 0; t < 2; ++t) {
    int n = lc + 16 * t;
    for (int f = 0; f < 8; ++f) {
      int k = 4 * f + 2 * half;
      float e0 = (n < HID && k     < HID) ? W_hh[n * HID + k]     : 0.0f;
      float e1 = (n < HID && k + 1 < HID) ? W_hh[n * HID + k + 1] : 0.0f;
      Bf[t][f] = (v2f){e0, e1};
    }
  }
  // W_out aligned with A-fragment K positions (for the y reduction)
  v2f Wo[8];
  for (int f = 0; f < 8; ++f) {
    int k = 4 * f + 2 * half;
    Wo[f] = (v2f){(k < HID) ? W_out[k] : 0.0f,
                  (k + 1 < HID) ? W_out[k + 1] : 0.0f};
  }
  // Bias and W_ih as per-lane (N-indexed) constants for the C layout
  float biasT[2], wihT[2];
  for (int t = 0; t < 2; ++t) {
    int n = lc + 16 * t;
    biasT[t] = (n < HID) ? (b_ih[n] + b_hh[n]) : 0.0f;
    wihT[t]  = (n < HID) ? W_ih[n] : 0.0f;
  }
  const float bout = b_out[0];

  // ---- Recurrence ------------------------------------------------------
  for (int s = 0; s < SEQ; ++s) {
    // A fragments = h_{s-1} in A layout: A(m,k), m=lc, k=4f+2*half+{0,1}
    v2f A[8];
#pragma unroll
    for (int f = 0; f < 8; ++f)
      A[f] = *(const v2f*)(hbuf + lc * LROW + 4 * f + 2 * half);

    // y[s-1] = h_{s-1} . W_out  (sum over K in-register, halves combined)
    if (s > 0) {
      float py = 0.0f;
#pragma unroll
      for (int f = 0; f < 8; ++f) py += A[f].x * Wo[f].x + A[f].y * Wo[f].y;
      py += swap_half(py);
      if (lane < 16) Y[(size_t)(s - 1) * BATCH + b0 + lc] = py + bout;
    }

    // Broadcast x loads: xm[r] = X[s, b0 + r + 8*half], r = 0..7
    const float* xp = X + (size_t)s * BATCH + b0 + 8 * half;
    v4f x0 = *(const v4f*)xp;
    v4f x1 = *(const v4f*)(xp + 4);
    if (s + 8 < SEQ) __builtin_prefetch(xp + (size_t)8 * BATCH, 0, 3);

    // Accumulators seeded with bias (constant over M -> replicate per VGPR)
    v8f acc0, acc1;
#pragma unroll
    for (int r = 0; r < 8; ++r) { acc0[r] = biasT[0]; acc1[r] = biasT[1]; }

    // h_{s-1} @ W_hh^T : 2 N-tiles x 8 K-fragments of 16x16x4 f32 WMMA
#pragma unroll
    for (int f = 0; f < 8; ++f) {
      acc0 = __builtin_amdgcn_wmma_f32_16x16x4_f32(
          false, A[f], false, Bf[0][f], (short)0, acc0, false, false);
      acc1 = __builtin_amdgcn_wmma_f32_16x16x4_f32(
          false, A[f], false, Bf[1][f], (short)0, acc1, false, false);
    }

    // h_s = tanh(acc + x * W_ih); store back in C layout:
    // element r -> (m = r + 8*half, n = lc + 16*t)
    float xm[8] = {x0.x, x0.y, x0.z, x0.w, x1.x, x1.y, x1.z, x1.w};
#pragma unroll
    for (int r = 0; r < 8; ++r) {
      float h0 = fast_tanh(acc0[r] + xm[r] * wihT[0]);
      float h1 = fast_tanh(acc1[r] + xm[r] * wihT[1]);
      int m = r + 8 * half;
      hbuf[m * LROW + lc]      = h0;
      hbuf[m * LROW + lc + 16] = h1;
    }
  }

  // Epilogue: y[SEQ-1] from h_{SEQ-1}
  {
    float py = 0.0f;
#pragma unroll
    for (int f = 0; f < 8; ++f) {
      v2f a = *(const v2f*)(hbuf + lc * LROW + 4 * f + 2 * half);
      py += a.x * Wo[f].x + a.y * Wo[f].y;
    }
    py += swap_half(py);
    if (lane < 16) Y[(size_t)(SEQ - 1) * BATCH + b0 + lc] = py + bout;
  }
}

extern "C" void kernel_launch(void* const* d_in, const int* in_sizes, int n_in,
                              void* d_out, int out_size, void* d_ws, size_t ws_size,
                              hipStream_t stream) {
  (void)in_sizes; (void)n_in; (void)out_size; (void)d_ws; (void)ws_size;
  const float* X     = (const float*)d_in[0];
  const float* W_ih  = (const float*)d_in[1];
  const float* W_hh  = (const float*)d_in[2];
  const float* b_ih  = (const float*)d_in[3];
  const float* b_hh  = (const float*)d_in[4];
  const float* W_out = (const float*)d_in[5];
  const float* b_out = (const float*)d_in[6];
  float* Y = (float*)d_out;

  dim3 grid(BATCH / 16 / WPB);   // 64 blocks
  dim3 block(32 * WPB);          // 4 wave32s
  hipLaunchKernelGGL(rnn_tanh_wmma_kernel, grid, block, 0, stream,
                     X, W_ih, W_hh, b_ih, b_hh, W_out, b_out, Y);
}